// SparseAttention_41712722378781
// MI455X (gfx1250) — compile-verified
//
#include <hip/hip_runtime.h>

typedef __attribute__((ext_vector_type(2))) float v2f;
typedef __attribute__((ext_vector_type(4))) float v4f;
typedef __attribute__((ext_vector_type(8))) float v8f;

#define B_  2
#define L_  2048
#define H_  16
#define E_  64
#define BS  256
#define NB  (L_ / BS)      // 8 query blocks
#define NTHREADS 256
#define NWAVES 8

// Dynamic LDS layout (327680 bytes total):
//   P  : float[256][256]  at float offset 0        (256 KB) -- softmax probs
//   KV : float[256][64]   at float offset 65536    ( 64 KB) -- K tile, then V tile
extern __shared__ float lds[];

__global__ __launch_bounds__(NTHREADS)
void blockdiag_attn_f32wmma(const float* __restrict__ Q,
                            const float* __restrict__ K,
                            const float* __restrict__ V,
                            float* __restrict__ out,    // [B, L, H, E]
                            float* __restrict__ attn)   // [B, H, L, L]
{
    float* Pl  = lds;              // 256*256
    float* KVl = lds + 256 * 256;  // 256*64

    const int bid = blockIdx.x;
    const int qb  = bid % NB;
    const int h   = (bid / NB) % H_;
    const int b   = bid / (NB * H_);

    const int tid    = threadIdx.x;
    const int wave   = tid >> 5;
    const int lane   = tid & 31;
    const int lane16 = lane & 15;
    const int khalf  = (lane < 16) ? 0 : 2;  // K-pair base within 4-wide chunk
    const int mhalf  = (lane < 16) ? 0 : 8;  // C/D row half

    const int q0 = qb * BS;  // query row base == key col base (diagonal block)

    // ---------- Phase 0: cooperative load of K tile [256 x 64] into LDS ----------
    {
        const float* Kbase = K + ((size_t)b * L_ + q0) * (H_ * E_) + (size_t)h * E_;
        #pragma unroll
        for (int i = 0; i < 16; ++i) {
            int idx = tid + i * NTHREADS;       // 4096 float4 total
            int row = idx >> 4;
            int f4  = idx & 15;
            v4f v = *(const v4f*)(Kbase + (size_t)row * (H_ * E_) + f4 * 4);
            *(v4f*)(KVl + row * E_ + f4 * 4) = v;
        }
    }
    __syncthreads();

    // ---------- Phase 1: S = Q K^T (f32 WMMA), row softmax, P -> LDS ----------
    for (int s = 0; s < 2; ++s) {
        const int strip = wave + s * NWAVES;            // 0..15
        const int qrow  = q0 + strip * 16 + lane16;     // row this lane supplies for A

        // A fragments for the whole 16x64 Q strip: 16 k-steps of float2
        v2f aq[16];
        const float* Qrow = Q + (((size_t)b * L_ + qrow) * H_ + h) * E_;
        #pragma unroll
        for (int kk = 0; kk < 16; ++kk)
            aq[kk] = *(const v2f*)(Qrow + kk * 4 + khalf);

        v8f acc[16];
        #pragma unroll
        for (int t = 0; t < 16; ++t) {                  // 16 N-tiles => 256 cols
            v8f c = {};
            const float* Kcol = KVl + (t * 16 + lane16) * E_ + khalf;
            #pragma unroll
            for (int kk = 0; kk < 16; ++kk) {           // K = 64 in steps of 4
                v2f bk = *(const v2f*)(Kcol + kk * 4);
                c = __builtin_amdgcn_wmma_f32_16x16x4_f32(
                        false, aq[kk], false, bk, (short)0, c, false, false);
            }
            acc[t] = c;
        }

        // Row softmax fully in registers: row M = v + mhalf lives in VGPR v across
        // the 16 lanes of one half-wave -> shfl_xor masks 1,2,4,8 stay in-row.
        #pragma unroll
        for (int v = 0; v < 8; ++v) {
            float m = -3.402823466e+38f;
            #pragma unroll
            for (int t = 0; t < 16; ++t) m = fmaxf(m, acc[t][v]);
            #pragma unroll
            for (int off = 1; off <= 8; off <<= 1)
                m = fmaxf(m, __shfl_xor(m, off, 32));

            float sum = 0.0f;
            #pragma unroll
            for (int t = 0; t < 16; ++t) {
                float e = __expf(acc[t][v] - m);
                acc[t][v] = e;
                sum += e;
            }
            #pragma unroll
            for (int off = 1; off <= 8; off <<= 1)
                sum += __shfl_xor(sum, off, 32);

            float r = 1.0f / sum;
            #pragma unroll
            for (int t = 0; t < 16; ++t) acc[t][v] *= r;
        }

        // Spill P strip to LDS (C-layout -> row-major)
        #pragma unroll
        for (int t = 0; t < 16; ++t) {
            #pragma unroll
            for (int v = 0; v < 8; ++v)
                Pl[(strip * 16 + v + mhalf) * 256 + t * 16 + lane16] = acc[t][v];
        }
    }
    __syncthreads();

    // ---------- Phase 2a: load V tile into the same LDS region ----------
    {
        const float* Vbase = V + ((size_t)b * L_ + q0) * (H_ * E_) + (size_t)h * E_;
        #pragma unroll
        for (int i = 0; i < 16; ++i) {
            int idx = tid + i * NTHREADS;
            int row = idx >> 4;
            int f4  = idx & 15;
            v4f v = *(const v4f*)(Vbase + (size_t)row * (H_ * E_) + f4 * 4);
            *(v4f*)(KVl + row * E_ + f4 * 4) = v;
        }
    }
    __syncthreads();

    // ---------- Phase 2b: O = P @ V (f32 WMMA over K=256) ----------
    for (int s = 0; s < 2; ++s) {
        const int strip = wave + s * NWAVES;
        v8f o[4] = {{}, {}, {}, {}};
        const float* Prow = Pl + (strip * 16 + lane16) * 256 + khalf;
        #pragma unroll 8
        for (int kk = 0; kk < 64; ++kk) {
            v2f a = *(const v2f*)(Prow + kk * 4);
            int k0 = kk * 4 + khalf;
            #pragma unroll
            for (int t = 0; t < 4; ++t) {               // 4 N-tiles => E = 64
                v2f bv;
                bv.x = KVl[(size_t)k0 * E_ + t * 16 + lane16];
                bv.y = KVl[(size_t)(k0 + 1) * E_ + t * 16 + lane16];
                o[t] = __builtin_amdgcn_wmma_f32_16x16x4_f32(
                           false, a, false, bv, (short)0, o[t], false, false);
            }
        }
        #pragma unroll
        for (int t = 0; t < 4; ++t) {
            #pragma unroll
            for (int v = 0; v < 8; ++v) {
                int r = q0 + strip * 16 + v + mhalf;
                out[(((size_t)b * L_ + r) * H_ + h) * E_ + t * 16 + lane16] = o[t][v];
            }
        }
    }

    // ---------- Phase 3: stream full 256 x 2048 attn row-strip (NT stores) ----------
    {
        float* Arow0 = attn + (((size_t)b * H_ + h) * L_ + q0) * L_;
        const int kbase = q0;
        for (int idx = tid; idx < 256 * 512; idx += NTHREADS) {  // float4 units
            int row = idx >> 9;
            int col = (idx & 511) * 4;
            v4f v = {};
            if ((unsigned)(col - kbase) < 256u)
                v = *(const v4f*)(Pl + row * 256 + (col - kbase));
            __builtin_nontemporal_store(v, (v4f*)(Arow0 + (size_t)row * L_ + col));
        }
    }
}

extern "C" void kernel_launch(void* const* d_in, const int* in_sizes, int n_in,
                              void* d_out, int out_size, void* d_ws, size_t ws_size,
                              hipStream_t stream) {
    const float* Q = (const float*)d_in[0];
    const float* K = (const float*)d_in[1];
    const float* V = (const float*)d_in[2];
    // d_in[3] = block_size scalar (== 256, compiled in)

    float* out  = (float*)d_out;                         // B*L*H*E = 4194304 floats
    float* attn = out + (size_t)B_ * L_ * H_ * E_;       // B*H*L*L = 134217728 floats

    const size_t shmem = (256 * 256 + 256 * 64) * sizeof(float);  // 327680 B
    static_assert((256 * 256 + 256 * 64) * sizeof(float) == 327680, "lds size");
    hipFuncSetAttribute((const void*)blockdiag_attn_f32wmma,
                        hipFuncAttributeMaxDynamicSharedMemorySize, (int)shmem);

    dim3 grid(B_ * H_ * NB);   // 256 workgroups
    dim3 block(NTHREADS);      // 8 waves (wave32)
    blockdiag_attn_f32wmma<<<grid, block, shmem, stream>>>(Q, K, V, out, attn);
}